// SpectralSpatialHamiltonianNet_28346784153890
// MI455X (gfx1250) — compile-verified
//
#include <hip/hip_runtime.h>
#include <math.h>

// ---- model dims ----
#define Bn   128
#define CINc 100
#define Pp   15
#define Lp   225          // P*P
#define Dm   64
#define Hh   8
#define DKk  8
#define Mrows (Bn*Lp)     // 28800 = 1800 * 16

typedef __attribute__((ext_vector_type(2))) float v2f;
typedef __attribute__((ext_vector_type(8))) float v8f;

__device__ __forceinline__ float gelu_f(float x) {
  return 0.5f * x * (1.0f + erff(x * 0.7071067811865476f));
}

// One K=4 step of V_WMMA_F32_16X16X4_F32.
// Lane (half = lane>>4, r = lane&15):
//   A: a0 = A[m0+r][k0+2*half], a1 = A[m0+r][k0+2*half+1]
//   B: b0 = B[k0+2*half][n0+r], b1 = B[k0+2*half+1][n0+r]
//   C/D: c[j] = C[m0 + j + 8*half][n0 + r]
__device__ __forceinline__ v8f wmma4(v8f c, float a0, float a1, float b0, float b1) {
  v2f a; a[0] = a0; a[1] = a1;
  v2f b; b[0] = b0; b[1] = b1;
  return __builtin_amdgcn_wmma_f32_16x16x4_f32(false, a, false, b, (short)0, c, false, false);
}

// ============================================================================
// Stage 1: conv1x1 100->64 (+BN eval +GELU).  A[m][k] = x[b][k][pix] (NCHW in).
// grid 1800 blocks x 128 thr (4 waves = 4 N-tiles of 16). Out NHWC [M,64].
// ============================================================================
__global__ void stage1_kernel(const float* __restrict__ X, const float* __restrict__ W,
                              const float* __restrict__ bc, const float* __restrict__ g,
                              const float* __restrict__ bnb, float* __restrict__ out) {
  const int K = CINc;
  int lane = threadIdx.x & 31, half = lane >> 4, r = lane & 15;
  int n0 = (threadIdx.x >> 5) * 16;
  int m0 = blockIdx.x * 16;
  int row = m0 + r, bidx = row / Lp, pix = row - bidx * Lp;
  const float* xb = X + (size_t)bidx * K * Lp + pix;
  v8f c = {};
  for (int k0 = 0; k0 < K; k0 += 4) {
    int ka = k0 + 2 * half;
    float a0 = xb[(size_t)ka * Lp];
    float a1 = xb[(size_t)(ka + 1) * Lp];
    float b0 = W[(n0 + r) * K + ka];
    float b1 = W[(n0 + r) * K + ka + 1];
    c = wmma4(c, a0, a1, b0, b1);
  }
  int n = n0 + r;
  float s = g[n] * rsqrtf(1.0f + 1e-5f);
  float bco = bc[n], bbo = bnb[n];
  for (int j = 0; j < 8; j++) {
    int orow = m0 + j + 8 * half;
    float t = (c[j] + bco) * s + bbo;
    out[(size_t)orow * Dm + n] = gelu_f(t);
  }
}

// ============================================================================
// Stage 2: conv1x1 64->64 (+BN+GELU). A row-major NHWC [M,64].
// ============================================================================
__global__ void stage2_kernel(const float* __restrict__ X, const float* __restrict__ W,
                              const float* __restrict__ bc, const float* __restrict__ g,
                              const float* __restrict__ bnb, float* __restrict__ out) {
  int lane = threadIdx.x & 31, half = lane >> 4, r = lane & 15;
  int n0 = (threadIdx.x >> 5) * 16;
  int m0 = blockIdx.x * 16;
  const float* arow = X + (size_t)(m0 + r) * Dm;
  v8f c = {};
  for (int k0 = 0; k0 < Dm; k0 += 4) {
    int ka = k0 + 2 * half;
    float a0 = arow[ka], a1 = arow[ka + 1];
    float b0 = W[(n0 + r) * Dm + ka], b1 = W[(n0 + r) * Dm + ka + 1];
    c = wmma4(c, a0, a1, b0, b1);
  }
  int n = n0 + r;
  float s = g[n] * rsqrtf(1.0f + 1e-5f);
  float bco = bc[n], bbo = bnb[n];
  for (int j = 0; j < 8; j++) {
    int orow = m0 + j + 8 * half;
    float t = (c[j] + bco) * s + bbo;
    out[(size_t)orow * Dm + n] = gelu_f(t);
  }
}

// ============================================================================
// Depthwise 3x3 'SAME' (+BN+GELU, optional +pos). NHWC layout, c fastest.
// ============================================================================
__global__ void dw_kernel(const float* __restrict__ in, float* __restrict__ out,
                          const float* __restrict__ W, const float* __restrict__ bc,
                          const float* __restrict__ g, const float* __restrict__ bnb,
                          const float* __restrict__ pos) {
  int idx = blockIdx.x * blockDim.x + threadIdx.x;
  if (idx >= Bn * Lp * Dm) return;
  int c = idx & 63;
  int l = (idx >> 6) % Lp;
  int b = idx / (Lp * Dm);
  int y = l / Pp, x = l - Pp * y;
  float acc = 0.0f;
  #pragma unroll
  for (int dy = -1; dy <= 1; dy++) {
    int yy = y + dy;
    if (yy < 0 || yy >= Pp) continue;
    #pragma unroll
    for (int dx = -1; dx <= 1; dx++) {
      int xx = x + dx;
      if (xx < 0 || xx >= Pp) continue;
      acc += W[c * 9 + (dy + 1) * 3 + (dx + 1)] *
             in[((size_t)b * Lp + (yy * Pp + xx)) * Dm + c];
    }
  }
  float s = g[c] * rsqrtf(1.0f + 1e-5f);
  float t = gelu_f((acc + bc[c]) * s + bnb[c]);
  if (pos) t += pos[l * Dm + c];
  out[idx] = t;
}

// ============================================================================
// Fused Q/P/K/V projection: grid (1800, 4mat) x 128 thr. No bias.
// ============================================================================
__global__ void qpkv_kernel(const float* __restrict__ Xseq,
                            const float* __restrict__ Wq, const float* __restrict__ Wp,
                            const float* __restrict__ Wk, const float* __restrict__ Wv,
                            float* __restrict__ qo, float* __restrict__ po,
                            float* __restrict__ ko, float* __restrict__ vo) {
  const float* W; float* O;
  switch (blockIdx.y) {
    case 0:  W = Wq; O = qo; break;
    case 1:  W = Wp; O = po; break;
    case 2:  W = Wk; O = ko; break;
    default: W = Wv; O = vo; break;
  }
  int lane = threadIdx.x & 31, half = lane >> 4, r = lane & 15;
  int n0 = (threadIdx.x >> 5) * 16;
  int m0 = blockIdx.x * 16;
  const float* arow = Xseq + (size_t)(m0 + r) * Dm;
  v8f c = {};
  for (int k0 = 0; k0 < Dm; k0 += 4) {
    int ka = k0 + 2 * half;
    float a0 = arow[ka], a1 = arow[ka + 1];
    float b0 = W[(n0 + r) * Dm + ka], b1 = W[(n0 + r) * Dm + ka + 1];
    c = wmma4(c, a0, a1, b0, b1);
  }
  int n = n0 + r;
  for (int j = 0; j < 8; j++)
    O[(size_t)(m0 + j + 8 * half) * Dm + n] = c[j];
}

// ============================================================================
// Symplectic attention, one block per (b,h). dk=8, L=225 in LDS.
// ============================================================================
__device__ void attn_softmax_av(const float* __restrict__ q,
                                const float (*__restrict__ Ks)[DKk],
                                const float (*__restrict__ Vs)[DKk],
                                float* __restrict__ outv) {
  const float scale = 0.35355339059327373f; // 1/sqrt(8)
  float mx = -3.0e38f;
  for (int j = 0; j < Lp; j++) {
    float s = 0.f;
    #pragma unroll
    for (int d = 0; d < DKk; d++) s += q[d] * Ks[j][d];
    mx = fmaxf(mx, s * scale);
  }
  float den = 0.f;
  float acc[DKk] = {0, 0, 0, 0, 0, 0, 0, 0};
  for (int j = 0; j < Lp; j++) {
    float s = 0.f;
    #pragma unroll
    for (int d = 0; d < DKk; d++) s += q[d] * Ks[j][d];
    float e = expf(s * scale - mx);
    den += e;
    #pragma unroll
    for (int d = 0; d < DKk; d++) acc[d] += e * Vs[j][d];
  }
  float inv = 1.0f / den;
  #pragma unroll
  for (int d = 0; d < DKk; d++) outv[d] = acc[d] * inv;
}

__global__ void attn_kernel(const float* __restrict__ qb, const float* __restrict__ pbuf,
                            const float* __restrict__ kb, const float* __restrict__ vb,
                            float* __restrict__ qn, float* __restrict__ pn,
                            float* __restrict__ pe,
                            const float* __restrict__ dtp, const float* __restrict__ gatep) {
  __shared__ float Ks[Lp][DKk], Vs[Lp][DKk], Qs[Lp][DKk], Ps[Lp][DKk];
  int b = blockIdx.x, h = blockIdx.y;
  const size_t base = (size_t)b * Lp * Dm + h * DKk;
  for (int idx = threadIdx.x; idx < Lp * DKk; idx += blockDim.x) {
    int l = idx >> 3, d = idx & 7;
    size_t gi = base + (size_t)l * Dm + d;
    Ks[l][d] = kb[gi];
    Vs[l][d] = vb[gi];
    Qs[l][d] = qb[gi];
    Ps[l][d] = pbuf[gi];
  }
  __syncthreads();
  float dt = dtp[0], gate = gatep[0];
  int i = threadIdx.x;
  if (i < Lp) {
    float q[DKk], phalf[DKk], pe1[DKk], pe2[DKk];
    #pragma unroll
    for (int d = 0; d < DKk; d++) q[d] = Qs[i][d];
    attn_softmax_av(q, Ks, Vs, pe1);
    #pragma unroll
    for (int d = 0; d < DKk; d++) {
      phalf[d] = Ps[i][d] - 0.5f * dt * pe1[d] * gate;
      q[d] = q[d] + dt * phalf[d];           // q_new
    }
    attn_softmax_av(q, Ks, Vs, pe2);
    #pragma unroll
    for (int d = 0; d < DKk; d++) {
      float pnew = phalf[d] - 0.5f * dt * pe2[d] * gate;
      size_t gi = base + (size_t)i * Dm + d;
      qn[gi] = q[d];
      pn[gi] = pnew;
      pe[gi] = 0.5f * (pe1[d] + pe2[d]);
    }
  }
}

// ============================================================================
// Output projection (q_new @ Wo^T + bo) + residual + LayerNorm.
// One wave computes a 16x64 stripe (4 WMMA N-tiles), LN via LDS. In-place safe.
// ============================================================================
__global__ void outln_kernel(const float* __restrict__ Qn, const float* __restrict__ resid,
                             const float* __restrict__ Wo, const float* __restrict__ bo,
                             const float* __restrict__ ln_g, const float* __restrict__ ln_b,
                             float* __restrict__ outp) {
  __shared__ float tile[16][68];
  int lane = threadIdx.x & 31, half = lane >> 4, r = lane & 15;
  int m0 = blockIdx.x * 16;
  const float* arow = Qn + (size_t)(m0 + r) * Dm;
  v8f acc[4];
  for (int t = 0; t < 4; t++) acc[t] = (v8f){};
  for (int k0 = 0; k0 < Dm; k0 += 4) {
    int ka = k0 + 2 * half;
    float a0 = arow[ka], a1 = arow[ka + 1];
    #pragma unroll
    for (int t = 0; t < 4; t++) {
      int n0 = t * 16;
      float b0 = Wo[(n0 + r) * Dm + ka], b1 = Wo[(n0 + r) * Dm + ka + 1];
      acc[t] = wmma4(acc[t], a0, a1, b0, b1);
    }
  }
  for (int t = 0; t < 4; t++) {
    int n = t * 16 + r;
    for (int j = 0; j < 8; j++) {
      int rl = j + 8 * half;
      tile[rl][n] = acc[t][j] + bo[n] + resid[(size_t)(m0 + rl) * Dm + n];
    }
  }
  __syncthreads();
  if (threadIdx.x < 16) {
    int rl = threadIdx.x;
    float mu = 0.f;
    for (int n = 0; n < Dm; n++) mu += tile[rl][n];
    mu *= (1.0f / Dm);
    float var = 0.f;
    for (int n = 0; n < Dm; n++) { float d = tile[rl][n] - mu; var += d * d; }
    var *= (1.0f / Dm);
    float inv = rsqrtf(var + 1e-5f);
    for (int n = 0; n < Dm; n++)
      outp[(size_t)(m0 + rl) * Dm + n] = (tile[rl][n] - mu) * inv * ln_g[n] + ln_b[n];
  }
}

// ============================================================================
// Head: pyramid (as weighted means) + global-context MLP + descriptor + cls.
// One 256-thread block per batch element.
// ============================================================================
__device__ __forceinline__ float cnt1d(int y, int pad) {
  int lo = y - pad; if (lo < 0) lo = 0;
  int hi = y + pad; if (hi > Pp - 1) hi = Pp - 1;
  return (float)(hi - lo + 1);
}

__global__ void head_kernel(
    const float* __restrict__ seq, const float* __restrict__ qn,
    const float* __restrict__ pn,  const float* __restrict__ pe,
    const float* __restrict__ msW0, const float* __restrict__ msW1,
    const float* __restrict__ msW2, const float* __restrict__ msW3,
    const float* __restrict__ msB0, const float* __restrict__ msB1,
    const float* __restrict__ msB2, const float* __restrict__ msB3,
    const float* __restrict__ gcW1, const float* __restrict__ gcB1,
    const float* __restrict__ gcW2, const float* __restrict__ gcB2,
    const float* __restrict__ qW1, const float* __restrict__ qB1,
    const float* __restrict__ qW2, const float* __restrict__ qB2,
    const float* __restrict__ pW1, const float* __restrict__ pB1,
    const float* __restrict__ pW2, const float* __restrict__ pB2,
    const float* __restrict__ vW1, const float* __restrict__ vB1,
    const float* __restrict__ vW2, const float* __restrict__ vB2,
    const float* __restrict__ fW,  const float* __restrict__ fB,
    const float* __restrict__ cW1, const float* __restrict__ cB1,
    const float* __restrict__ cW2, const float* __restrict__ cB2,
    float* __restrict__ out) {
  int b = blockIdx.x, tid = threadIdx.x;
  __shared__ float sw[4][64];
  __shared__ float qmean[64], qvar[64], vmean[64], vstd[64], md[64];
  __shared__ float red[256];
  __shared__ float scal4[4];
  __shared__ float t16a[16], t16b[16], t16c[16];
  __shared__ float comb8[8], desc8[8];
  __shared__ float msv[64], gh[32], g16[16];
  __shared__ float comb88[88], h1v[64];
  const size_t boff = (size_t)b * Lp * Dm;

  if (tid < 64) {
    int c = tid;
    float s1 = 0, s3 = 0, s5 = 0, s7 = 0, sq = 0, sqq = 0, sv = 0, svv = 0, sp = 0;
    for (int l = 0; l < Lp; l++) {
      int y = l / Pp, xx = l - Pp * y;
      size_t idx = boff + (size_t)l * Dm + c;
      float fs = seq[idx];
      s1 += fs;
      s3 += fs * cnt1d(y, 1) * cnt1d(xx, 1);
      s5 += fs * cnt1d(y, 2) * cnt1d(xx, 2);
      s7 += fs * cnt1d(y, 3) * cnt1d(xx, 3);
      float fq = qn[idx]; sq += fq; sqq += fq * fq;
      float fv = pe[idx]; sv += fv; svv += fv * fv;
      sp += pn[idx];
    }
    sw[0][c] = s1 * (1.f / 225.f);
    sw[1][c] = s3 * (1.f / (225.f * 9.f));
    sw[2][c] = s5 * (1.f / (225.f * 25.f));
    sw[3][c] = s7 * (1.f / (225.f * 49.f));
    qmean[c] = sq * (1.f / 225.f);
    qvar[c]  = (sqq - sq * sq * (1.f / 225.f)) * (1.f / 224.f);
    vmean[c] = sv * (1.f / 225.f);
    float vv = (svv - sv * sv * (1.f / 225.f)) * (1.f / 224.f);
    vstd[c]  = sqrtf(fmaxf(vv, 0.f));
    md[c]    = sp * (1.f / 225.f);
  }
  __syncthreads();

  // ||p||_2 per token, then mean/std over tokens (ddof=1)
  float pl = 0.f;
  if (tid < Lp) {
    float acc = 0.f;
    for (int c2 = 0; c2 < 64; c2++) {
      float v = pn[boff + (size_t)tid * Dm + c2];
      acc += v * v;
    }
    pl = sqrtf(acc);
  }
  red[tid] = pl; __syncthreads();
  for (int s = 128; s > 0; s >>= 1) { if (tid < s) red[tid] += red[tid + s]; __syncthreads(); }
  float sumP = red[0]; __syncthreads();
  red[tid] = pl * pl; __syncthreads();
  for (int s = 128; s > 0; s >>= 1) { if (tid < s) red[tid] += red[tid + s]; __syncthreads(); }
  float sumP2 = red[0]; __syncthreads();
  if (tid == 0) {
    float ke   = 0.5f * sumP * (1.f / 225.f);
    float pstd = sqrtf(fmaxf((sumP2 - sumP * sumP * (1.f / 225.f)) * (1.f / 224.f), 0.f));
    float smd = 0, smd2 = 0;
    for (int c2 = 0; c2 < 64; c2++) { smd += md[c2]; smd2 += md[c2] * md[c2]; }
    float mm = sqrtf(smd2);
    float mv = (smd2 - smd * smd * (1.f / 64.f)) * (1.f / 63.f);
    scal4[0] = ke; scal4[1] = mm; scal4[2] = pstd; scal4[3] = mv;
  }
  __syncthreads();

  // first-layer matvecs, disjoint thread ranges
  if (tid < 64) {                      // pyramid: 4 x (16x64)
    int i = tid >> 4, o = tid & 15;
    const float* Wp_ = (i == 0) ? msW0 : (i == 1) ? msW1 : (i == 2) ? msW2 : msW3;
    const float* Bp_ = (i == 0) ? msB0 : (i == 1) ? msB1 : (i == 2) ? msB2 : msB3;
    float acc = Bp_[o];
    for (int c2 = 0; c2 < 64; c2++) acc += Wp_[o * 64 + c2] * sw[i][c2];
    msv[tid] = acc;
  } else if (tid < 96) {               // gc layer1: 32x64
    int o = tid - 64; float acc = gcB1[o];
    for (int c2 = 0; c2 < 64; c2++) acc += gcW1[o * 64 + c2] * sw[0][c2];
    gh[o] = gelu_f(acc);
  } else if (tid < 112) {              // q-desc hidden: 16x128
    int j = tid - 96; float acc = qB1[j];
    for (int c2 = 0; c2 < 64; c2++) acc += qW1[j * 128 + c2] * qmean[c2];
    for (int c2 = 0; c2 < 64; c2++) acc += qW1[j * 128 + 64 + c2] * qvar[c2];
    t16a[j] = gelu_f(acc);
  } else if (tid < 128) {              // v-desc hidden: 16x128
    int j = tid - 112; float acc = vB1[j];
    for (int c2 = 0; c2 < 64; c2++) acc += vW1[j * 128 + c2] * vmean[c2];
    for (int c2 = 0; c2 < 64; c2++) acc += vW1[j * 128 + 64 + c2] * vstd[c2];
    t16c[j] = gelu_f(acc);
  } else if (tid < 144) {              // p-desc hidden: 16x4
    int j = tid - 128; float acc = pB1[j];
    for (int m2 = 0; m2 < 4; m2++) acc += pW1[j * 4 + m2] * scal4[m2];
    t16b[j] = gelu_f(acc);
  }
  __syncthreads();

  if (tid < 16) {                      // gc layer2: 16x32
    float acc = gcB2[tid];
    for (int j = 0; j < 32; j++) acc += gcW2[tid * 32 + j] * gh[j];
    g16[tid] = acc;
  } else if (tid < 18) {               // q-desc out: 2x16
    int o = tid - 16; float acc = qB2[o];
    for (int j = 0; j < 16; j++) acc += qW2[o * 16 + j] * t16a[j];
    comb8[o] = acc;
  } else if (tid < 20) {               // p-desc out: 2x16
    int o = tid - 18; float acc = pB2[o];
    for (int j = 0; j < 16; j++) acc += pW2[o * 16 + j] * t16b[j];
    comb8[2 + o] = acc;
  } else if (tid < 24) {               // v-desc out: 4x16
    int o = tid - 20; float acc = vB2[o];
    for (int j = 0; j < 16; j++) acc += vW2[o * 16 + j] * t16c[j];
    comb8[4 + o] = acc;
  }
  __syncthreads();
  if (tid < 8) {                       // descriptor fusion: 8x8 + gelu
    float acc = fB[tid];
    for (int j = 0; j < 8; j++) acc += fW[tid * 8 + j] * comb8[j];
    desc8[tid] = gelu_f(acc);
  }
  __syncthreads();
  if (tid < 64)       comb88[tid] = msv[tid];
  else if (tid < 80)  comb88[tid] = g16[tid - 64];
  else if (tid < 88)  comb88[tid] = desc8[tid - 80];
  __syncthreads();
  if (tid < 64) {                      // classifier hidden: 64x88 + gelu
    float acc = cB1[tid];
    for (int j = 0; j < 88; j++) acc += cW1[tid * 88 + j] * comb88[j];
    h1v[tid] = gelu_f(acc);
  }
  __syncthreads();
  if (tid < 16) {                      // logits: 16x64
    float acc = cB2[tid];
    for (int j = 0; j < 64; j++) acc += cW2[tid * 64 + j] * h1v[j];
    out[b * 16 + tid] = acc;
  }
}

// ============================================================================
// Launcher. Input index map (setup_inputs insertion order, depth-first):
//  0:x 1:se1_W 2:se1_b 3:bn1_g 4:bn1_b 5:se2_W 6:se2_b 7:bn2_g 8:bn2_b
//  9:dw1_W 10:dw1_b 11:bn3_g 12:bn3_b 13:dw2_W 14:dw2_b 15:bn4_g 16:bn4_b
//  17:pos  18+10*l: Wq Wp Wk Wv Wo bo ln_g ln_b dt gate (l=0..3)
//  58-61:ms_W[4] 62-65:ms_b[4] 66:gc_W1 67:gc_b1 68:gc_W2 69:gc_b2
//  70-73:desc.q 74-77:desc.p 78-81:desc.v 82:fW 83:fb
//  84:cls_W1 85:cls_b1 86:cls_W2 87:cls_b2
// ============================================================================
extern "C" void kernel_launch(void* const* d_in, const int* in_sizes, int n_in,
                              void* d_out, int out_size, void* d_ws, size_t ws_size,
                              hipStream_t stream) {
  (void)in_sizes; (void)n_in; (void)out_size; (void)ws_size;
  auto in = [&](int i) { return (const float*)d_in[i]; };
  float* ws = (float*)d_ws;
  const size_t NM = (size_t)Mrows * Dm;     // 1,843,200 floats per activation
  float* h1  = ws + 0 * NM;
  float* h2  = ws + 1 * NM;
  float* h3  = ws + 2 * NM;
  float* seq = ws + 3 * NM;
  float* qb  = ws + 4 * NM;
  float* pb  = ws + 5 * NM;
  float* kb  = ws + 6 * NM;
  float* vb  = ws + 7 * NM;
  float* qnb = ws + 8 * NM;
  float* pnb = ws + 9 * NM;
  float* peb = ws + 10 * NM;

  const int MT = Mrows / 16;                // 1800 row tiles
  stage1_kernel<<<dim3(MT), 128, 0, stream>>>(in(0), in(1), in(2), in(3), in(4), h1);
  stage2_kernel<<<dim3(MT), 128, 0, stream>>>(h1, in(5), in(6), in(7), in(8), h2);
  const int tot = Bn * Lp * Dm;
  dw_kernel<<<(tot + 255) / 256, 256, 0, stream>>>(h2, h3, in(9), in(10), in(11), in(12), nullptr);
  dw_kernel<<<(tot + 255) / 256, 256, 0, stream>>>(h3, seq, in(13), in(14), in(15), in(16), in(17));

  for (int l = 0; l < 4; l++) {
    int base = 18 + 10 * l;
    qpkv_kernel<<<dim3(MT, 4), 128, 0, stream>>>(seq, in(base + 0), in(base + 1),
                                                 in(base + 2), in(base + 3),
                                                 qb, pb, kb, vb);
    attn_kernel<<<dim3(Bn, Hh), 256, 0, stream>>>(qb, pb, kb, vb, qnb, pnb, peb,
                                                  in(base + 8), in(base + 9));
    outln_kernel<<<dim3(MT), 32, 0, stream>>>(qnb, seq, in(base + 4), in(base + 5),
                                              in(base + 6), in(base + 7), seq);
  }

  head_kernel<<<dim3(Bn), 256, 0, stream>>>(
      seq, qnb, pnb, peb,
      in(58), in(59), in(60), in(61), in(62), in(63), in(64), in(65),
      in(66), in(67), in(68), in(69),
      in(70), in(71), in(72), in(73),
      in(74), in(75), in(76), in(77),
      in(78), in(79), in(80), in(81),
      in(82), in(83),
      in(84), in(85), in(86), in(87),
      (float*)d_out);
}